// SpikingDlPallium_48842368090334
// MI455X (gfx1250) — compile-verified
//
#include <hip/hip_runtime.h>
#include <stdint.h>

#define N_DG 128
#define N_CA3 4096
#define N_CA1 1024
#define MAX_EP 8192
#define EPSF 1e-8f

typedef float v2f __attribute__((ext_vector_type(2)));
typedef float v4f __attribute__((ext_vector_type(4)));
typedef float v8f __attribute__((ext_vector_type(8)));

// D = A(16x4 f32) * B(4x16 f32) + C(16x16 f32), full f32 precision WMMA.
__device__ __forceinline__ v8f wmma4(v2f a, v2f b, v8f c) {
  return __builtin_amdgcn_wmma_f32_16x16x4_f32(false, a, false, b, (short)0, c,
                                               false, false);
}

// Izhikevich RS run, T steps, noise is [T][stride] row-major.
__device__ __forceinline__ float izh_run(float drive, const float* noise,
                                         int stride, int idx, float tonic,
                                         int T) {
  float v = -65.0f, u = 0.2f * -65.0f, r = 0.0f;
  for (int t = 0; t < T; ++t) {
    float I = drive + noise[t * stride + idx] + tonic;
    float vn = v + (0.04f * v * v + 5.0f * v + 140.0f - u + I);
    float un = u + 0.02f * (0.2f * v - u);
    float sp = (vn >= 30.0f) ? 1.0f : 0.0f;
    v = (sp > 0.0f) ? -65.0f : vn;
    u = un + sp * 8.0f;
    r = 0.9f * r + 0.1f * sp;
  }
  return r;
}

// ---- K0: zero the cross-kernel accumulators in workspace ------------------
__global__ void k0_init(float* acc, unsigned long long* best) {
  acc[0] = 0.f;  // qnorm2
  acc[1] = 0.f;  // dot(rate_ca3, last)
  acc[2] = 0.f;  // |rate_ca3|^2
  acc[3] = 0.f;  // |last|^2
  acc[4] = 0.f;  // pattern_separation
  *best = 0ull;
}

// ---- K1: DG Izhikevich (128 neurons, 8 steps) + pattern separation --------
__global__ void k1_dg(const float* place, const float* noise_dg, float* out_rate,
                      float* acc) {
  int i = threadIdx.x;
  float r = izh_run(place[i] * 25.0f, noise_dg, N_DG, i, -1.5f, 8);
  out_rate[i] = r;
  __shared__ float s[N_DG];
  s[i] = (r > 0.1f) ? 1.0f : 0.0f;
  __syncthreads();
  for (int off = 64; off; off >>= 1) {
    if (i < off) s[i] += s[i + off];
    __syncthreads();
  }
  if (i == 0) acc[4] = 1.0f - s[0] * (1.0f / 128.0f);
}

// ---- K2: W_dg_ca3 @ {rate_dg, dg_drive} (one weight read, two vectors) ----
__global__ void k2_dg_gemv(const float* __restrict__ W, const float* place,
                           const float* rate_dg, float* mossy, float* qraw,
                           float* acc) {
  __shared__ float sr[N_DG], sd[N_DG];
  int tid = threadIdx.x;
  if (tid < N_DG) {
    sr[tid] = rate_dg[tid];
    sd[tid] = place[tid] * 25.0f;
  }
  __syncthreads();
  int l = tid & 31, w = tid >> 5;
  int row = blockIdx.x * 8 + w;
  v4f av = __builtin_nontemporal_load(
      reinterpret_cast<const v4f*>(W + (size_t)row * N_DG + l * 4));
  int k = l * 4;
  float m = av.x * sr[k] + av.y * sr[k + 1] + av.z * sr[k + 2] + av.w * sr[k + 3];
  float q = av.x * sd[k] + av.y * sd[k + 1] + av.z * sd[k + 2] + av.w * sd[k + 3];
  for (int off = 16; off; off >>= 1) {
    m += __shfl_xor(m, off, 32);
    q += __shfl_xor(q, off, 32);
  }
  if (l == 0) {
    mossy[row] = m;
    qraw[row] = q;
    atomicAdd(&acc[0], q * q);
  }
}

// ---- K3: recurrent WMMA-GEMV + theta mix + CA3 Izhikevich + cos accum -----
// Weights streamed non-temporally (one use per call); x vector LDS-resident.
__global__ void k3_rec(const float* __restrict__ Wrec, const float* prev,
                       const float* mossy, const float* theta,
                       const float* noise_ca3, const float* last,
                       float* out_rate_ca3, float* acc) {
  __shared__ float xv[N_CA3];
  __shared__ float part[8][16];
  int tid = threadIdx.x;
  for (int i = tid; i < N_CA3 / 4; i += 256)
    reinterpret_cast<float4*>(xv)[i] = reinterpret_cast<const float4*>(prev)[i];
  __syncthreads();

  int l = tid & 31, w = tid >> 5;
  int row16 = l & 15, half = l >> 4;
  float bmask = (row16 == 0) ? 1.0f : 0.0f;  // only column 0 of B is live
  int grow = blockIdx.x * 16 + row16;
  const float* arow = Wrec + (size_t)grow * N_CA3;
  int kb0 = w * (N_CA3 / 8);
  v8f c0 = {}, c1 = {};
#pragma unroll 4
  for (int kb = kb0; kb < kb0 + N_CA3 / 8; kb += 8) {
    v2f a0 = __builtin_nontemporal_load(
        reinterpret_cast<const v2f*>(arow + kb + 2 * half));
    v2f x0 = *reinterpret_cast<const v2f*>(xv + kb + 2 * half);
    c0 = wmma4(a0, x0 * bmask, c0);
    v2f a1 = __builtin_nontemporal_load(
        reinterpret_cast<const v2f*>(arow + kb + 4 + 2 * half));
    v2f x1 = *reinterpret_cast<const v2f*>(xv + kb + 4 + 2 * half);
    c1 = wmma4(a1, x1 * bmask, c1);
  }
  v8f c = c0 + c1;
  if (l == 0) {
#pragma unroll
    for (int j = 0; j < 8; ++j) part[w][j] = c[j];
  } else if (l == 16) {
#pragma unroll
    for (int j = 0; j < 8; ++j) part[w][8 + j] = c[j];
  }
  __syncthreads();
  if (tid < 16) {
    float dot = 0.f;
#pragma unroll
    for (int w2 = 0; w2 < 8; ++w2) dot += part[w2][tid];
    int r = blockIdx.x * 16 + tid;
    float tm = sinf(theta[0]);
    float enc = fmaxf(0.f, tm) * 0.7f + 0.3f;
    float ret = fmaxf(0.f, -tm) * 0.7f + 0.3f;
    float I = mossy[r] * enc * 15.0f + dot * ret * 3.0f;
    float rr = izh_run(I, noise_ca3, N_CA3, r, -0.5f, 20);
    out_rate_ca3[r] = rr;
    float lv = last[r];
    atomicAdd(&acc[1], rr * lv);
    atomicAdd(&acc[2], rr * rr);
    atomicAdd(&acc[3], lv * lv);
  }
}

// ---- K5: episodes WMMA-GEMV + per-row norm + global argmax ----------------
// Episodes use default (RT) caching: 128 MB stays resident in the 192 MB L2
// across graph replays while the NT-streamed weights bypass it.
__global__ void k5_ep(const float* __restrict__ ep, const float* qraw,
                      const float* acc, unsigned long long* best) {
  __shared__ float q[N_CA3];
  __shared__ float pqd[8][16];
  __shared__ float pss[8][16];
  int tid = threadIdx.x;
  for (int i = tid; i < N_CA3 / 4; i += 256)
    reinterpret_cast<float4*>(q)[i] = reinterpret_cast<const float4*>(qraw)[i];
  __syncthreads();

  int l = tid & 31, w = tid >> 5;
  int row16 = l & 15, half = l >> 4;
  float bmask = (row16 == 0) ? 1.0f : 0.0f;
  int grow = blockIdx.x * 16 + row16;
  const float* arow = ep + (size_t)grow * N_CA3;
  int kb0 = w * (N_CA3 / 8);
  v8f c0 = {}, c1 = {};
  float ssl = 0.f;  // per-row self-dot side accumulator (VALU co-exec, free)
#pragma unroll 4
  for (int kb = kb0; kb < kb0 + N_CA3 / 8; kb += 8) {
    v2f a0 = *reinterpret_cast<const v2f*>(arow + kb + 2 * half);
    v2f x0 = *reinterpret_cast<const v2f*>(q + kb + 2 * half);
    c0 = wmma4(a0, x0 * bmask, c0);
    ssl += a0.x * a0.x + a0.y * a0.y;
    v2f a1 = *reinterpret_cast<const v2f*>(arow + kb + 4 + 2 * half);
    v2f x1 = *reinterpret_cast<const v2f*>(q + kb + 4 + 2 * half);
    c1 = wmma4(a1, x1 * bmask, c1);
    ssl += a1.x * a1.x + a1.y * a1.y;
  }
  v8f c = c0 + c1;
  float ss_tot = ssl + __shfl_xor(ssl, 16, 32);  // pair (l, l+16) -> full row
  if (l == 0) {
#pragma unroll
    for (int j = 0; j < 8; ++j) pqd[w][j] = c[j];
  } else if (l == 16) {
#pragma unroll
    for (int j = 0; j < 8; ++j) pqd[w][8 + j] = c[j];
  }
  if (l < 16) pss[w][l] = ss_tot;
  __syncthreads();

  if (tid < 32) {  // wave 0 does the 16-row argmax
    float sim;
    int row;
    if (tid < 16) {
      float qd = 0.f, ss = 0.f;
#pragma unroll
      for (int w2 = 0; w2 < 8; ++w2) {
        qd += pqd[w2][tid];
        ss += pss[w2][tid];
      }
      float qn = sqrtf(acc[0]);
      float qs = (qn > EPSF) ? 1.0f / (qn + EPSF) : 1.0f;
      sim = qd * qs / (sqrtf(ss) + EPSF);
      row = blockIdx.x * 16 + tid;
    } else {
      sim = -3.0e38f;
      row = 0x7FFFFFFF;
    }
#pragma unroll
    for (int m = 8; m >= 1; m >>= 1) {
      float os = __shfl_xor(sim, m, 32);
      int orow = __shfl_xor(row, m, 32);
      if (os > sim || (os == sim && orow < row)) {
        sim = os;
        row = orow;
      }
    }
    if (tid == 0) {
      unsigned bits = __float_as_uint(sim);
      unsigned key = (bits & 0x80000000u) ? ~bits : (bits | 0x80000000u);
      unsigned long long packed =
          ((unsigned long long)key << 32) | (unsigned)(0x7FFFFFFF - row);
      atomicMax(best, packed);
    }
  }
}

// ---- K6: CA1 WMMA-GEMV + context gate + CA1 Izhikevich --------------------
__global__ void k6_ca1(const float* __restrict__ Wc, const float* rate_ca3,
                       const float* ctx, const float* noise_ca1,
                       float* out_rate_ca1) {
  __shared__ float xv[N_CA3];
  __shared__ float part[8][16];
  int tid = threadIdx.x;
  for (int i = tid; i < N_CA3 / 4; i += 256)
    reinterpret_cast<float4*>(xv)[i] =
        reinterpret_cast<const float4*>(rate_ca3)[i];
  __syncthreads();

  int l = tid & 31, w = tid >> 5;
  int row16 = l & 15, half = l >> 4;
  float bmask = (row16 == 0) ? 1.0f : 0.0f;
  int grow = blockIdx.x * 16 + row16;
  const float* arow = Wc + (size_t)grow * N_CA3;
  int kb0 = w * (N_CA3 / 8);
  v8f c0 = {}, c1 = {};
#pragma unroll 4
  for (int kb = kb0; kb < kb0 + N_CA3 / 8; kb += 8) {
    v2f a0 = __builtin_nontemporal_load(
        reinterpret_cast<const v2f*>(arow + kb + 2 * half));
    v2f x0 = *reinterpret_cast<const v2f*>(xv + kb + 2 * half);
    c0 = wmma4(a0, x0 * bmask, c0);
    v2f a1 = __builtin_nontemporal_load(
        reinterpret_cast<const v2f*>(arow + kb + 4 + 2 * half));
    v2f x1 = *reinterpret_cast<const v2f*>(xv + kb + 4 + 2 * half);
    c1 = wmma4(a1, x1 * bmask, c1);
  }
  v8f c = c0 + c1;
  if (l == 0) {
#pragma unroll
    for (int j = 0; j < 8; ++j) part[w][j] = c[j];
  } else if (l == 16) {
#pragma unroll
    for (int j = 0; j < 8; ++j) part[w][8 + j] = c[j];
  }
  __syncthreads();
  if (tid < 16) {
    float dot = 0.f;
#pragma unroll
    for (int w2 = 0; w2 < 8; ++w2) dot += part[w2][tid];
    int r = blockIdx.x * 16 + tid;
    float cg = 1.0f + 0.3f * 0.25f * (ctx[0] + ctx[1] + ctx[2] + ctx[3]);
    float rr = izh_run(dot * cg, noise_ca1, N_CA1, r, -0.5f, 8);
    out_rate_ca1[r] = rr;
  }
}

// ---- K7: scalar epilogue ---------------------------------------------------
__global__ void k7_scalars(const float* ctx, const float* ep_out,
                           const float* ep_str, const float* acc,
                           const unsigned long long* best, float* outs) {
  // pattern completion
  float n_ca3 = sqrtf(acc[2]);
  float n_last = sqrtf(acc[3]);
  float cs = acc[1] / (n_ca3 * n_last + EPSF);
  float valid = (n_ca3 > EPSF && n_last > EPSF) ? 1.0f : 0.0f;
  float pat_comp = fmaxf(0.0f, cs) * valid;
  // episodic retrieval
  unsigned long long p = *best;
  unsigned key = (unsigned)(p >> 32);
  unsigned bits = (key & 0x80000000u) ? (key ^ 0x80000000u) : ~key;
  float bsim = __uint_as_float(bits);
  int idx = 0x7FFFFFFF - (int)(unsigned)(p & 0xFFFFFFFFu);
  idx = min(max(idx, 0), MAX_EP - 1);
  float gate = (bsim > 0.4f) ? 1.0f : 0.0f;
  float o0 = ep_out[2 * idx], o1 = ep_out[2 * idx + 1];
  float st = ep_str[idx];
  float food = o0 * bsim * st * gate;
  float risk = o1 * bsim * st * gate;
  float novelty = (bsim > 0.4f) ? fmaxf(0.0f, 1.0f - bsim) : 1.0f;
  // predictive coding column (pe = [0, ctx[2], 0, 0])
  float c2 = ctx[2];
  outs[0] = food;
  outs[1] = risk;
  outs[2] = food - risk;
  outs[3] = novelty;
  outs[4] = acc[4];
  outs[5] = pat_comp;
  outs[6] = fabsf(c2) * 0.25f;
  outs[7] = 1.0f;
  outs[8] = 0.5f * c2 * c2;
}

extern "C" void kernel_launch(void* const* d_in, const int* in_sizes, int n_in,
                              void* d_out, int out_size, void* d_ws,
                              size_t ws_size, hipStream_t stream) {
  const float* place = (const float*)d_in[0];
  const float* ctx = (const float*)d_in[1];
  const float* theta = (const float*)d_in[2];
  const float* last = (const float*)d_in[3];
  const float* prev = (const float*)d_in[4];
  const float* W_dg_ca3 = (const float*)d_in[5];
  const float* W_rec = (const float*)d_in[6];
  const float* W_ca1 = (const float*)d_in[7];
  const float* ep_place = (const float*)d_in[8];
  const float* ep_out = (const float*)d_in[9];
  const float* ep_str = (const float*)d_in[10];
  const float* noise_dg = (const float*)d_in[11];
  const float* noise_ca3 = (const float*)d_in[12];
  const float* noise_ca1 = (const float*)d_in[13];

  float* out = (float*)d_out;
  float* out_dg = out;                         // [128]
  float* out_ca3 = out + N_DG;                 // [4096]
  float* out_ca1 = out + N_DG + N_CA3;         // [1024]
  float* out_sc = out + N_DG + N_CA3 + N_CA1;  // [9]

  float* ws = (float*)d_ws;
  float* mossy = ws;            // [4096]
  float* qraw = ws + N_CA3;     // [4096]
  float* acc = ws + 2 * N_CA3;  // [5] accumulators
  unsigned long long* best =
      (unsigned long long*)(ws + 2 * N_CA3 + 6);  // 8B aligned

  k0_init<<<1, 1, 0, stream>>>(acc, best);
  k1_dg<<<1, 128, 0, stream>>>(place, noise_dg, out_dg, acc);
  k2_dg_gemv<<<N_CA3 / 8, 256, 0, stream>>>(W_dg_ca3, place, out_dg, mossy,
                                            qraw, acc);
  k3_rec<<<N_CA3 / 16, 256, 0, stream>>>(W_rec, prev, mossy, theta, noise_ca3,
                                         last, out_ca3, acc);
  k5_ep<<<MAX_EP / 16, 256, 0, stream>>>(ep_place, qraw, acc, best);
  k6_ca1<<<N_CA1 / 16, 256, 0, stream>>>(W_ca1, out_ca3, ctx, noise_ca1,
                                         out_ca1);
  k7_scalars<<<1, 1, 0, stream>>>(ctx, ep_out, ep_str, acc, best, out_sc);
}